// IDTCWT_28097676051076
// MI455X (gfx1250) — compile-verified
//
#include <hip/hip_runtime.h>

typedef __attribute__((ext_vector_type(2))) float v2f;
typedef __attribute__((ext_vector_type(8))) float v8f;

#define INV_SQRT2F 0.70710678118654752440f

// Compile-time banded synthesis-filter matrices.
// out[p0+i] = sum_r M[i][r] * in[(j0+r) mod Nin], j0 = p0/2 - 2, support r in [0,12).
// Synthesis taps f[m] = AF[...][9-m] (time-reversed analysis filters).
struct BandTab {
  float m[2][2][2][16][12];  // [bank][tree][lp|hp][row i][r]
  constexpr BandTab() : m() {
    const float AF[2][2][2][10] = {
      { // bank 0: Farras analysis (first stage)
        { { 0.0f, -0.08838834764832f, 0.08838834764832f, 0.69587998903400f, 0.69587998903400f,
            0.08838834764832f, -0.08838834764832f, 0.01122679215254f, 0.01122679215254f, 0.0f },
          { 0.0f, -0.01122679215254f, 0.01122679215254f, 0.08838834764832f, 0.08838834764832f,
            -0.69587998903400f, 0.69587998903400f, -0.08838834764832f, -0.08838834764832f, 0.0f } },
        { { 0.01122679215254f, 0.01122679215254f, -0.08838834764832f, 0.08838834764832f, 0.69587998903400f,
            0.69587998903400f, 0.08838834764832f, -0.08838834764832f, 0.0f, 0.0f },
          { 0.0f, 0.0f, -0.08838834764832f, -0.08838834764832f, 0.69587998903400f,
            -0.69587998903400f, 0.08838834764832f, 0.08838834764832f, 0.01122679215254f, -0.01122679215254f } } },
      { // bank 1: Kingsbury Q-shift analysis
        { { 0.03516384f, 0.0f, -0.08832942f, 0.23389032f, 0.76027237f, 0.58751830f, 0.0f, -0.11430184f, 0.0f, 0.0f },
          { 0.0f, 0.0f, -0.11430184f, 0.0f, 0.58751830f, -0.76027237f, 0.23389032f, 0.08832942f, 0.0f, -0.03516384f } },
        { { 0.0f, 0.0f, -0.11430184f, 0.0f, 0.58751830f, 0.76027237f, 0.23389032f, -0.08832942f, 0.0f, 0.03516384f },
          { -0.03516384f, 0.0f, 0.08832942f, 0.23389032f, -0.76027237f, 0.58751830f, 0.0f, -0.11430184f, 0.0f, 0.0f } } }
    };
    for (int b = 0; b < 2; ++b)
      for (int t = 0; t < 2; ++t)
        for (int h = 0; h < 2; ++h)
          for (int i = 0; i < 16; ++i)
            for (int r = 0; r < 12; ++r) {
              float v = 0.0f;
              if ((i & 1) == 0) {
                int e = (i >> 1) + 4 - r;               // taps f[2e]
                if (e >= 0 && e <= 4) v = AF[b][t][h][9 - 2 * e];
              } else {
                int e = ((i + 3) >> 1) + 2 - r;         // taps f[2e+1]
                if (e >= 0 && e <= 4) v = AF[b][t][h][9 - (2 * e + 1)];
              }
              m[b][t][h][i][r] = v;
            }
  }
};
__constant__ BandTab g_band{};

// Load one K-block B fragment: spatial positions j, j+1 (periodic), channel = lane%16.
// MIX: value = (x0 + sign*x1) * INV_SQRT2 (fused dual-tree butterfly).
template <int MIX>
__device__ __forceinline__ v2f load_block(const float* __restrict__ p0_,
                                          const float* __restrict__ p1_,
                                          float sgn, int in_base, int j,
                                          int nin_mask, int sp, int chan) {
  const int jA = (j + nin_mask + 1) & nin_mask;
  const int jB = (j + 1 + nin_mask + 1) & nin_mask;
  const int offA = in_base + jA * sp + chan;
  const int offB = in_base + jB * sp + chan;
  v2f b;
  b.x = p0_[offA];
  b.y = p0_[offB];
  if (MIX) {
    b.x = (b.x + sgn * p1_[offA]) * INV_SQRT2F;
    b.y = (b.y + sgn * p1_[offB]) * INV_SQRT2F;
  }
  return b;
}

// Generic 1-D synthesis filter bank along an arbitrary axis.
// One wave computes 4 consecutive 16(position) x 16(channel) output tiles; each tile is
// 6 chained V_WMMA_F32_16X16X4_F32 (K=12 lo + K=12 hi, single f32 accumulator).
// K-block 2 of tile t == K-block 0 of tile t+1 -> carried in registers (25% fewer loads).
template <int MIX_LO, int MIX_HI, int ACCUM>
__global__ __launch_bounds__(256) void sfb1d_wmma(
    const float* __restrict__ lo0, const float* __restrict__ lo1,
    const float* __restrict__ hi0, const float* __restrict__ hi1,
    float mix_sign, float* __restrict__ out, float out_scale,
    int nin_mask,            // n_in - 1 (n_in power of two)
    int grp_shift,           // log2(groups of 64 outputs per line)
    int n_lines,
    int in_inner_mask, int in_inner_shift, int in_ostride, int in_istride, int in_spstride,
    int out_inner_mask, int out_inner_shift, int out_ostride, int out_istride, int out_spstride,
    int bank, int tree)
{
  const int lane = threadIdx.x & 31;
  const int wave = blockIdx.x * (blockDim.x >> 5) + (threadIdx.x >> 5);
  const int line = wave >> grp_shift;
  if (line >= n_lines) return;               // wave-uniform exit (EXEC all-1 for WMMA)
  const int grp = wave & ((1 << grp_shift) - 1);

  const int chan = lane & 15;                // N index (channel) and A-matrix row M
  const int half = lane >> 4;                // K offsets {0,1} vs {2,3}
  const int kb   = half << 1;

  // A fragments: constant banded filter matrices (built at compile time), scale folded.
  const float* __restrict__ Ml = g_band.m[bank][tree][0][chan];
  const float* __restrict__ Mh = g_band.m[bank][tree][1][chan];
  v2f a_lo[3], a_hi[3];
#pragma unroll
  for (int s = 0; s < 3; ++s) {
    const int k0 = 4 * s + kb;
    a_lo[s].x = out_scale * Ml[k0];
    a_lo[s].y = out_scale * Ml[k0 + 1];
    a_hi[s].x = out_scale * Mh[k0];
    a_hi[s].y = out_scale * Mh[k0 + 1];
  }

  const int in_base  = (line >> in_inner_shift) * in_ostride  + (line & in_inner_mask) * in_istride;
  const int out_base = (line >> out_inner_shift) * out_ostride + (line & out_inner_mask) * out_istride;

  const int jbase = (grp << 5) - 2;          // j0 of tile 0; j0(t) = jbase + 8*t

  // Software pipeline: preload the 3 K-blocks of tile 0.
  v2f bl[3], bh[3];
#pragma unroll
  for (int s = 0; s < 3; ++s) {
    bl[s] = load_block<MIX_LO>(lo0, lo1, mix_sign, in_base, jbase + 4 * s + kb,
                               nin_mask, in_spstride, chan);
    bh[s] = load_block<MIX_HI>(hi0, hi1, mix_sign, in_base, jbase + 4 * s + kb,
                               nin_mask, in_spstride, chan);
  }

#pragma unroll
  for (int t = 0; t < 4; ++t) {
    const int p0 = (grp << 6) + (t << 4);
    v8f acc = {0.f, 0.f, 0.f, 0.f, 0.f, 0.f, 0.f, 0.f};

#pragma unroll
    for (int s = 0; s < 3; ++s) {
      acc = __builtin_amdgcn_wmma_f32_16x16x4_f32(false, a_lo[s], false, bl[s],
                                                  (short)0, acc, false, false);
      acc = __builtin_amdgcn_wmma_f32_16x16x4_f32(false, a_hi[s], false, bh[s],
                                                  (short)0, acc, false, false);
    }

    // Advance the pipeline: block 2 becomes block 0; fetch blocks 1,2 of tile t+1.
    if (t < 3) {
      const int jn = jbase + 8 * (t + 1);
      bl[0] = bl[2];
      bh[0] = bh[2];
      bl[1] = load_block<MIX_LO>(lo0, lo1, mix_sign, in_base, jn + 4 + kb,
                                 nin_mask, in_spstride, chan);
      bh[1] = load_block<MIX_HI>(hi0, hi1, mix_sign, in_base, jn + 4 + kb,
                                 nin_mask, in_spstride, chan);
      bl[2] = load_block<MIX_LO>(lo0, lo1, mix_sign, in_base, jn + 8 + kb,
                                 nin_mask, in_spstride, chan);
      bh[2] = load_block<MIX_HI>(hi0, hi1, mix_sign, in_base, jn + 8 + kb,
                                 nin_mask, in_spstride, chan);
    }

    // D layout: VGPR r -> row p0 + r (lanes 0-15) / p0 + 8 + r (lanes 16-31)
#pragma unroll
    for (int r = 0; r < 8; ++r) {
      const int row = p0 + r + (half << 3);
      const int off = out_base + row * out_spstride + chan;
      float v = acc[r];
      if (ACCUM) v += out[off];
      out[off] = v;
    }
  }
}

static inline int ilog2i(int x) { int s = 0; while ((1 << s) < x) ++s; return s; }

static void launch_sfb(hipStream_t s,
                       const float* lo0, const float* lo1, const float* hi0, const float* hi1,
                       float sgn, float* outp, int accum, float scale,
                       int n_in, int n_lines,
                       int ii, int ios, int iis, int isp,
                       int oi, int oos, int ois, int osp,
                       int bank, int tree)
{
  const int grp_shift = ilog2i(n_in) - 5;          // (2*n_in)/64 groups per line
  const long long waves = (long long)n_lines << grp_shift;
  const int blocks = (int)((waves + 7) / 8);       // 8 waves (256 threads) per block
  const int nin_mask = n_in - 1;
  const int iim = ii - 1, iish = ilog2i(ii);
  const int oim = oi - 1, oish = ilog2i(oi);

#define SFB_ARGS lo0, lo1, hi0, hi1, sgn, outp, scale, nin_mask, grp_shift, n_lines, \
                 iim, iish, ios, iis, isp, oim, oish, oos, ois, osp, bank, tree
  if (lo1 != nullptr)            sfb1d_wmma<1, 1, 0><<<blocks, 256, 0, s>>>(SFB_ARGS);
  else if (hi1 != nullptr)       sfb1d_wmma<0, 1, 0><<<blocks, 256, 0, s>>>(SFB_ARGS);
  else if (accum)                sfb1d_wmma<0, 0, 1><<<blocks, 256, 0, s>>>(SFB_ARGS);
  else                           sfb1d_wmma<0, 0, 0><<<blocks, 256, 0, s>>>(SFB_ARGS);
#undef SFB_ARGS
}

extern "C" void kernel_launch(void* const* d_in, const int* in_sizes, int n_in_args,
                              void* d_out, int out_size, void* d_ws, size_t ws_size,
                              hipStream_t stream) {
  (void)in_sizes; (void)n_in_args; (void)out_size; (void)ws_size;
  const float* w1 = (const float*)d_in[0];   // [2,2,3,4,256,256,16]
  const float* w2 = (const float*)d_in[1];   // [2,2,3,4,128,128,16]
  const float* lo = (const float*)d_in[2];   // [2,2,4,128,128,16]
  float* out = (float*)d_out;                // [4,512,512,16]
  float* ws = (float*)d_ws;

  const long long SUB_W1 = 4LL * 256 * 256 * 16;
  const long long SUB_W2 = 4LL * 128 * 128 * 16;

  // Workspace: reused across the 4 branches (~100 MB total)
  float* lo2 = ws;                            // [4,128,256,16]
  float* h2  = lo2 + 4LL * 128 * 256 * 16;    // [4,128,256,16]
  float* x2  = h2  + 4LL * 128 * 256 * 16;    // [4,256,256,16]
  float* lo1 = x2  + 4LL * 256 * 256 * 16;    // [4,256,512,16]
  float* h1  = lo1 + 4LL * 256 * 512 * 16;    // [4,256,512,16]

  int bi = 0;
  for (int m = 0; m < 2; ++m)
  for (int n = 0; n < 2; ++n, ++bi) {
    // Butterfly mix: (m==n) pairs trees (0,0)&(1,1); else (0,1)&(1,0); sign=+ for m==0
    const float sgn = (m == 0) ? 1.0f : -1.0f;
    const int fa = (m == n) ? 0 : 1;   // flattened (mm*2+nn) of first pair member
    const int fb = (m == n) ? 3 : 2;   // flattened second pair member
    const float* lo_mn = lo + (long long)(m * 2 + n) * SUB_W2;
    auto W2P = [&](int flat, int o) { return w2 + ((long long)flat * 3 + o) * SUB_W2; };
    auto W1P = [&](int flat, int o) { return w1 + ((long long)flat * 3 + o) * SUB_W1; };

    // ---- Q-shift stage: [4,128,128,16] -> [4,256,256,16] ----
    // K1: W-axis, lo path + mixed LH          (bank 1, tree n)
    launch_sfb(stream, lo_mn, nullptr, W2P(fa, 0), W2P(fb, 0), sgn,
               lo2, 0, 1.0f, 128, 4 * 128,
               512, 0, 2048, 16,   512, 0, 4096, 16,   1, n);
    // K2: W-axis, mixed HL + mixed HH
    launch_sfb(stream, W2P(fa, 1), W2P(fb, 1), W2P(fa, 2), W2P(fb, 2), sgn,
               h2, 0, 1.0f, 128, 4 * 128,
               512, 0, 2048, 16,   512, 0, 4096, 16,   1, n);
    // K3: H-axis combine                      (bank 1, tree m)
    launch_sfb(stream, lo2, nullptr, h2, nullptr, 0.f,
               x2, 0, 1.0f, 128, 4 * 256,
               256, 128 * 256 * 16, 16, 256 * 16,
               256, 256 * 256 * 16, 16, 256 * 16,   1, m);

    // ---- Farras stage: [4,256,256,16] -> [4,512,512,16] ----
    // K4: W-axis, x2 + mixed w1 LH            (bank 0, tree n)
    launch_sfb(stream, x2, nullptr, W1P(fa, 0), W1P(fb, 0), sgn,
               lo1, 0, 1.0f, 256, 4 * 256,
               1024, 0, 4096, 16,   1024, 0, 8192, 16,   0, n);
    // K5: W-axis, mixed w1 HL + mixed w1 HH
    launch_sfb(stream, W1P(fa, 1), W1P(fb, 1), W1P(fa, 2), W1P(fb, 2), sgn,
               h1, 0, 1.0f, 256, 4 * 256,
               1024, 0, 4096, 16,   1024, 0, 8192, 16,   0, n);
    // K6: H-axis combine, scale 0.5, accumulate into d_out  (bank 0, tree m)
    launch_sfb(stream, lo1, nullptr, h1, nullptr, 0.f,
               out, (bi > 0) ? 1 : 0, 0.5f, 256, 4 * 512,
               512, 256 * 512 * 16, 16, 512 * 16,
               512, 512 * 512 * 16, 16, 512 * 16,   0, m);
  }
}